// DownsampleMultiscale_32916629357225
// MI455X (gfx1250) — compile-verified
//
#include <hip/hip_runtime.h>
#include <hip/hip_bf16.h>
#include <stdint.h>

typedef __attribute__((ext_vector_type(2))) float v2f;
typedef __attribute__((ext_vector_type(8))) float v8f;

#define K_TOTAL    65536
#define ROWS_LO    1024
#define ROWS_HI    4096
#define ROWS_TOTAL 5120
#define KSPLIT     16
#define KW         (K_TOTAL / KSPLIT)   // 4096 K per wave-unit
#define BK         64                   // staged K chunk
#define NCHUNK     (KW / BK)            // 64 chunks (even -> clean 2x unroll)
#define A_STRIDE   (BK + 4)             // 68: 68%64==4 -> conflict-free column reads, 16B-aligned rows
#define WAVES      4                    // waves per workgroup (128 threads, wave32)

__global__ void zero_out_kernel(float* __restrict__ out, int n) {
    int i = blockIdx.x * blockDim.x + threadIdx.x;
    if (i < n) out[i] = 0.0f;
}

__global__ __launch_bounds__(128) void downsample_wmma_f32(
    const float* __restrict__ x,     // (2, 65536)
    const float* __restrict__ Xlo,   // (1024, 65536)
    const float* __restrict__ Xhi,   // (4096, 65536)
    float* __restrict__ out)         // [y_lo (2*1024) | y_hi (2*4096)]
{
    // Double-buffered, per-wave-private LDS tiles (no cross-wave sharing -> no barriers).
    __shared__ float As[WAVES][2][16][A_STRIDE]; // A tile: 16 rows x BK
    __shared__ float Bs[WAVES][2][3][BK];        // x chunk: ch0, ch1, and a zero row for N>=2 lanes

    const int lane = threadIdx.x & 31;
    const int wave = threadIdx.x >> 5;
    const int unit = blockIdx.x * WAVES + wave;     // 0..5119
    const int tile = unit >> 4;                     // 0..319  (row tile)
    const int ks   = unit & (KSPLIT - 1);           // 0..15   (K slice)
    const int row0 = tile * 16;
    const size_t kg0 = (size_t)ks * KW;

    const float* Xbase = (row0 < ROWS_LO)
        ? (Xlo + (size_t)row0 * K_TOTAL)
        : (Xhi + (size_t)(row0 - ROWS_LO) * K_TOTAL);

    // Zero the dummy "channel 2" row in both buffers (read by lanes with N>=2).
    Bs[wave][0][2][lane]      = 0.0f;
    Bs[wave][0][2][lane + 32] = 0.0f;
    Bs[wave][1][2][lane]      = 0.0f;
    Bs[wave][1][2][lane + 32] = 0.0f;

    // ---- async staging addresses -------------------------------------------
    // A tile chunk (16 x BK f32): 8 iters x 32 lanes of b128;
    // lane covers row r = 2*it + (lane>>4), float4-col = lane&15 (BK/4 == 16).
    const int r0 = lane >> 4;       // 0 or 1
    const int c0 = lane & 15;       // float4 column
    const float* gA = Xbase + (size_t)r0 * K_TOTAL + kg0 + (size_t)c0 * 4;
    const float* gB = x     + (size_t)r0 * K_TOTAL + kg0 + (size_t)c0 * 4; // ch = r0
    unsigned ldsA[2], ldsB[2];
    ldsA[0] = (unsigned)(uintptr_t)&As[wave][0][r0][c0 * 4];
    ldsA[1] = (unsigned)(uintptr_t)&As[wave][1][r0][c0 * 4];
    ldsB[0] = (unsigned)(uintptr_t)&Bs[wave][0][r0][c0 * 4];
    ldsB[1] = (unsigned)(uintptr_t)&Bs[wave][1][r0][c0 * 4];

    // 9 async b128 loads per chunk: 8 for A (16x64 f32), 1 for B (2x64 f32).
    auto prefetch = [&](int buf, int kofs) {
        #pragma unroll
        for (int it = 0; it < 8; ++it) {
            asm volatile("global_load_async_to_lds_b128 %0, %1, off"
                         :: "v"(ldsA[buf] + it * (2 * A_STRIDE * 4)),
                            "v"((uint64_t)(uintptr_t)(gA + (size_t)it * 2 * K_TOTAL + kofs))
                         : "memory");
        }
        asm volatile("global_load_async_to_lds_b128 %0, %1, off"
                     :: "v"(ldsB[buf]),
                        "v"((uint64_t)(uintptr_t)(gB + kofs))
                     : "memory");
    };

    // ---- fp32 WMMA fragment coordinates (16x4 A / 4x16 B, wave32) ----------
    //   lanes 0-15 : M/N = lane,     VGPR0 = K+0, VGPR1 = K+1
    //   lanes 16-31: M/N = lane-16,  VGPR0 = K+2, VGPR1 = K+3
    const int m    = lane & 15;
    const int koff = (lane >> 4) << 1;
    const int ch2  = (m < 2) ? m : 2;    // N>=2 lanes read the zero row (no cndmask)

    v8f acc = {};

    // `buf` must be a literal at each call site so the shared-array indexing
    // keeps addrspace(3) and lowers to ds_load_2addr_b64 (not flat_load).
    auto compute = [&](int buf) {
        #pragma unroll
        for (int kk = 0; kk < BK; kk += 4) {
            v2f a, b;
            a.x = As[wave][buf][m][kk + koff];
            a.y = As[wave][buf][m][kk + koff + 1];
            b.x = Bs[wave][buf][ch2][kk + koff];
            b.y = Bs[wave][buf][ch2][kk + koff + 1];
            acc = __builtin_amdgcn_wmma_f32_16x16x4_f32(
                /*neg_a=*/false, a, /*neg_b=*/false, b,
                /*c_mod=*/(short)0, acc, /*reuse_a=*/false, /*reuse_b=*/false);
        }
    };

    prefetch(0, 0);
    for (int chunk = 0; chunk < NCHUNK; chunk += 2) {
        // ---- buffer 0 ------------------------------------------------------
        asm volatile("s_wait_dscnt 0x0" ::: "memory");   // WAR: buf1 reads done
        prefetch(1, (chunk + 1) * BK);                   // chunk+1 <= 63 always
        asm volatile("s_wait_asynccnt 0x9" ::: "memory"); // buf0's 9 are done
        compute(0);

        // ---- buffer 1 ------------------------------------------------------
        asm volatile("s_wait_dscnt 0x0" ::: "memory");   // WAR: buf0 reads done
        if (chunk + 2 < NCHUNK) {
            prefetch(0, (chunk + 2) * BK);
            asm volatile("s_wait_asynccnt 0x9" ::: "memory");
        } else {
            asm volatile("s_wait_asynccnt 0x0" ::: "memory");
        }
        compute(1);
    }

    // ---- combine K-split partials: lanes with N<2 hold channel N -----------
    if (m < 2) {
        #pragma unroll
        for (int v = 0; v < 8; ++v) {
            int mrow = v + ((lane >> 4) << 3);   // C/D layout: VGPR v -> M = v (+8 upper half)
            int row  = row0 + mrow;
            int oidx = (row < ROWS_LO)
                ? (m * ROWS_LO + row)
                : (2 * ROWS_LO + m * ROWS_HI + (row - ROWS_LO));
            atomicAdd(&out[oidx], acc[v]);
        }
    }
}

extern "C" void kernel_launch(void* const* d_in, const int* in_sizes, int n_in,
                              void* d_out, int out_size, void* d_ws, size_t ws_size,
                              hipStream_t stream) {
    const float* x   = (const float*)d_in[0];
    const float* Xlo = (const float*)d_in[1];
    const float* Xhi = (const float*)d_in[2];
    float* out = (float*)d_out;

    // zero the accumulation target (harness poisons d_out; atomics need zeros)
    zero_out_kernel<<<(out_size + 255) / 256, 256, 0, stream>>>(out, out_size);

    // 5120 wave-units / 4 waves per block = 1280 blocks of 128 threads
    const int blocks = (ROWS_TOTAL / 16) * KSPLIT / WAVES;
    downsample_wmma_f32<<<blocks, 128, 0, stream>>>(x, Xlo, Xhi, out);
}